// TransformerMonotonic_18794776887972
// MI455X (gfx1250) — compile-verified
//
#include <hip/hip_runtime.h>
#include <hip/hip_bf16.h>

typedef __attribute__((ext_vector_type(2))) float v2f;
typedef __attribute__((ext_vector_type(8))) float v8f;
typedef unsigned int u32;

constexpr int L = 6, D = 512, H = 8, FF = 2048, S = 2048, T = 512, B = 2, DH = 64;
constexpr float SCALE = 0.125f;       // 1/sqrt(64)
constexpr float EPS = 1e-10f;
constexpr float LN_EPS = 1e-5f;

// ---------------- WMMA helper: exact fp32 16x16x4 matrix FMA --------------
__device__ __forceinline__ v8f wmma4(v2f a, v2f b, v8f c) {
  return __builtin_amdgcn_wmma_f32_16x16x4_f32(false, a, false, b, (short)0, c,
                                               false, false);
}

// ---------------- CDNA5 async global->LDS copy (ASYNCcnt tracked) ---------
// LDS byte address = low 32 bits of the generic pointer (ISA 10.2: for the
// LDS aperture).
__device__ __forceinline__ void cp_async16(const float* gsrc, float* ldst) {
  u32 lo = (u32)(unsigned long long)ldst;
  asm volatile("global_load_async_to_lds_b128 %0, %1, off"
               :: "v"(lo), "v"(gsrc)
               : "memory");
}
__device__ __forceinline__ void async_wait0() {
  asm volatile("s_wait_asynccnt 0x0" ::: "memory");
}

// ---------------- elementwise helpers ------------------------------------
__global__ void k_add(float* __restrict__ o, const float* __restrict__ a,
                      const float* __restrict__ b, int n) {
  int i = blockIdx.x * 256 + threadIdx.x;
  if (i < n) o[i] = b ? a[i] + b[i] : a[i];
}

// [Sdim,B,D] (seq-first) -> [B,Sdim,D], optional add of second seq-first input
__global__ void k_tr_sb2bs(float* __restrict__ o, const float* __restrict__ a,
                           const float* __restrict__ b, int Sdim) {
  int i = blockIdx.x * 256 + threadIdx.x;
  int total = B * Sdim * D;
  if (i >= total) return;
  int d = i % D;
  int s = (i / D) % Sdim;
  int bb = i / (D * Sdim);
  int src = (s * B + bb) * D + d;
  o[i] = b ? a[src] + b[src] : a[src];
}

// [B,Sdim,D] -> [Sdim,B,D]
__global__ void k_tr_bs2sb(float* __restrict__ o, const float* __restrict__ a,
                           int Sdim) {
  int i = blockIdx.x * 256 + threadIdx.x;
  int total = B * Sdim * D;
  if (i >= total) return;
  int d = i % D;
  int bb = (i / D) % B;
  int s = i / (D * B);
  o[i] = a[(bb * Sdim + s) * D + d];
}

// ---------------- generic GEMM: C[M,N] = A[M,K] * W[N,K]^T + bias ---------
// Block = 8 waves = 128(M) x 64(N) tile.  K consumed in 32-wide chunks that
// are staged into LDS with async global->LDS DMA, double buffered so the DMA
// of chunk c+1 overlaps the 32 WMMAs of chunk c.  Requires M%128==0,
// N%64==0, K%32==0 (true for every GEMM in this model).
constexpr int KC = 32;      // k-chunk (floats)
constexpr int KP = KC + 4;  // padded LDS row stride: 16B aligned, conflict-free
__global__ __launch_bounds__(256) void k_gemm(const float* __restrict__ A,
                                              const float* __restrict__ W,
                                              const float* __restrict__ bias,
                                              float* __restrict__ C, int M,
                                              int N, int K, int relu) {
  __shared__ float lA[2][128 * KP];
  __shared__ float lB[2][64 * KP];
  int tid = threadIdx.x, lane = tid & 31, wave = tid >> 5;
  int l16 = lane & 15, kh = (lane >> 4) * 2;
  int mblk = blockIdx.y * 128, n0 = blockIdx.x * 64;

  auto load_chunk = [&](int kc, int buf) {
    // A tile: 128 x 32 floats = 1024 float4 slots, 4 per thread
#pragma unroll
    for (int i = 0; i < 4; ++i) {
      int slot = tid + i * 256;
      int row = slot >> 3, c4 = (slot & 7) * 4;
      cp_async16(A + (size_t)(mblk + row) * K + kc + c4,
                 &lA[buf][row * KP + c4]);
    }
    // W tile: 64 x 32 floats = 512 float4 slots, 2 per thread
#pragma unroll
    for (int i = 0; i < 2; ++i) {
      int slot = tid + i * 256;
      int row = slot >> 3, c4 = (slot & 7) * 4;
      cp_async16(W + (size_t)(n0 + row) * K + kc + c4,
                 &lB[buf][row * KP + c4]);
    }
  };

  v8f acc[4] = {};
  load_chunk(0, 0);
  async_wait0();
  __syncthreads();
  int nch = K / KC;
  for (int c = 0; c < nch; ++c) {
    int cur = c & 1;
    if (c + 1 < nch) load_chunk((c + 1) * KC, cur ^ 1);
    const float* la = &lA[cur][(wave * 16 + l16) * KP + kh];
    const float* lb = &lB[cur][l16 * KP + kh];
#pragma unroll
    for (int k = 0; k < KC; k += 4) {
      v2f a = *(const v2f*)(la + k);
#pragma unroll
      for (int j = 0; j < 4; ++j) {
        v2f b = *(const v2f*)(lb + j * 16 * KP + k);
        acc[j] = wmma4(a, b, acc[j]);
      }
    }
    async_wait0();    // next chunk's DMA done (this wave's share)
    __syncthreads();  // everyone done with 'cur', DMA visible to all
  }

  int mrow = mblk + wave * 16 + (lane >> 4) * 8;
#pragma unroll
  for (int j = 0; j < 4; ++j) {
    int n = n0 + j * 16 + l16;
    float bv = bias ? bias[n] : 0.f;
#pragma unroll
    for (int r = 0; r < 8; ++r) {
      float v = acc[j][r] + bv;
      if (relu) v = fmaxf(v, 0.f);
      C[(size_t)(mrow + r) * N + n] = v;
    }
  }
}

// ---------------- fused self-attention ------------------------------------
// Q,K,V,O row layout: [(seq)*B + b, D], head slice h*64..  One block per
// (16-query tile, head, batch).  16 x Sk score strip kept entirely in LDS.
__global__ __launch_bounds__(256) void k_attn(const float* __restrict__ Q,
                                              const float* __restrict__ Km,
                                              const float* __restrict__ V,
                                              float* __restrict__ O, int Sk) {
  extern __shared__ float smem[];
  float* sc = smem;                    // 16 * Sk
  float* red = smem + (size_t)16 * Sk; // 256
  float* rmx = red + 256;              // 16
  float* rsm = rmx + 16;               // 16
  float* op = rsm + 16;                // 2 * 16 * 64

  int tid = threadIdx.x, lane = tid & 31, wave = tid >> 5;
  int l16 = lane & 15, kh = (lane >> 4) * 2;
  int q0 = blockIdx.x * 16, h = blockIdx.y, bb = blockIdx.z;
  size_t hoff = (size_t)h * DH;

  // preload the wave's Q fragments (whole 16x64 tile, 16 k-steps)
  const float* qrow = Q + ((size_t)(q0 + l16) * B + bb) * D + hoff + kh;
  v2f aq[16];
#pragma unroll
  for (int i = 0; i < 16; ++i) {
    aq[i].x = qrow[i * 4];
    aq[i].y = qrow[i * 4 + 1];
  }

  // ---- phase 1: scores = scale * Q K^T  (waves split the key range) ----
  int kb = Sk >> 3;  // keys per wave
  for (int kt = wave * kb; kt < (wave + 1) * kb; kt += 16) {
    v8f acc = {};
    const float* krow = Km + ((size_t)(kt + l16) * B + bb) * D + hoff + kh;
#pragma unroll
    for (int i = 0; i < 16; ++i) {
      v2f bf;
      bf.x = krow[i * 4];
      bf.y = krow[i * 4 + 1];
      acc = wmma4(aq[i], bf, acc);
    }
    int rbase = (lane >> 4) * 8;
#pragma unroll
    for (int r = 0; r < 8; ++r)
      sc[(size_t)(rbase + r) * Sk + kt + l16] = acc[r] * SCALE;
  }
  __syncthreads();

  // ---- phase 2: exact softmax rows in LDS (keep rows unnormalized) ----
  int row = tid >> 4, sub = tid & 15, chunk = Sk >> 4;
  float* srow = sc + (size_t)row * Sk + sub * chunk;
  float mx = -3.4e38f;
  for (int i = 0; i < chunk; ++i) mx = fmaxf(mx, srow[i]);
  red[tid] = mx;
  __syncthreads();
  if (sub == 0) {
    float m2 = red[tid];
    for (int i = 1; i < 16; ++i) m2 = fmaxf(m2, red[tid + i]);
    rmx[row] = m2;
  }
  __syncthreads();
  float rm = rmx[row];
  float sm = 0.f;
  for (int i = 0; i < chunk; ++i) {
    float e = expf(srow[i] - rm);
    srow[i] = e;
    sm += e;
  }
  red[tid] = sm;
  __syncthreads();
  if (sub == 0) {
    float s2 = 0.f;
    for (int i = 0; i < 16; ++i) s2 += red[tid + i];
    rsm[row] = s2;
  }
  __syncthreads();

  // ---- phase 3: O = P V  (waves: 4 n-subtiles x 2 key halves) ----
  int j = wave & 3, half = wave >> 2;
  int kbeg = half * (Sk >> 1), kend = kbeg + (Sk >> 1);
  v8f acc = {};
  for (int k = kbeg; k < kend; k += 4) {
    v2f a;
    a.x = sc[(size_t)l16 * Sk + k + kh];
    a.y = sc[(size_t)l16 * Sk + k + kh + 1];
    const float* v0 = V + ((size_t)(k + kh) * B + bb) * D + hoff + j * 16 + l16;
    v2f bf;
    bf.x = v0[0];
    bf.y = v0[(size_t)B * D];
    acc = wmma4(a, bf, acc);
  }
  int rbase = (lane >> 4) * 8;
#pragma unroll
  for (int r = 0; r < 8; ++r)
    op[(size_t)half * 1024 + (rbase + r) * 64 + j * 16 + l16] = acc[r];
  __syncthreads();

  for (int e = tid * 4; e < tid * 4 + 4; ++e) {
    int rr = e >> 6, nn = e & 63;
    float val = (op[rr * 64 + nn] + op[1024 + rr * 64 + nn]) / rsm[rr];
    O[((size_t)(q0 + rr) * B + bb) * D + hoff + nn] = val;
  }
}

// ---------------- MoChA score kernels -------------------------------------
// out[((b*H+h)*T + t)*S + s] = f(scale * q . k)
// mode 0: sigmoid(x + mr)   mode 1: raw x
__global__ __launch_bounds__(32) void k_qk(const float* __restrict__ Qh,
                                           const float* __restrict__ Kh,
                                           float* __restrict__ out,
                                           const float* __restrict__ mrp,
                                           int lidx, int mode) {
  int lane = threadIdx.x & 31, l16 = lane & 15, kh = (lane >> 4) * 2;
  int s0 = blockIdx.x * 64, t0 = blockIdx.y * 16, bh = blockIdx.z;
  int bb = bh / H, h = bh % H;
  const float* qrow = Qh + ((size_t)bb * T + t0 + l16) * D + h * DH + kh;
  v8f acc[4] = {};
#pragma unroll
  for (int i = 0; i < 16; ++i) {
    v2f a;
    a.x = qrow[i * 4];
    a.y = qrow[i * 4 + 1];
#pragma unroll
    for (int j = 0; j < 4; ++j) {
      const float* krow =
          Kh + ((size_t)bb * S + s0 + j * 16 + l16) * D + h * DH + kh;
      v2f bf;
      bf.x = krow[i * 4];
      bf.y = krow[i * 4 + 1];
      acc[j] = wmma4(a, bf, acc[j]);
    }
  }
  float mr = mrp ? mrp[lidx] : 0.f;
  int rbase = (lane >> 4) * 8;
#pragma unroll
  for (int j = 0; j < 4; ++j)
    for (int r = 0; r < 8; ++r) {
      float v = acc[j][r] * SCALE;
      if (mode == 0) v = 1.f / (1.f + expf(-(v + mr)));
      out[((size_t)bh * T + t0 + rbase + r) * S + s0 + j * 16 + l16] = v;
    }
}

// per row: eu = exp(u - rowmax); csum = inclusive cumsum(eu). u overwritten.
__global__ __launch_bounds__(256) void k_expcumsum(float* __restrict__ u,
                                                   float* __restrict__ csum) {
  __shared__ float lds[256];
  size_t row = blockIdx.x;
  float* ur = u + row * (size_t)S;
  float* cr = csum + row * (size_t)S;
  int tid = threadIdx.x;
  float loc[8];
  float mx = -3.4e38f;
#pragma unroll
  for (int i = 0; i < 8; ++i) {
    loc[i] = ur[tid * 8 + i];
    mx = fmaxf(mx, loc[i]);
  }
  lds[tid] = mx;
  __syncthreads();
  for (int off = 128; off > 0; off >>= 1) {
    if (tid < off) lds[tid] = fmaxf(lds[tid], lds[tid + off]);
    __syncthreads();
  }
  float rm = lds[0];
  __syncthreads();
  float run = 0.f;
#pragma unroll
  for (int i = 0; i < 8; ++i) {
    loc[i] = expf(loc[i] - rm);
    run += loc[i];
  }
  lds[tid] = run;
  __syncthreads();
  for (int off = 1; off < 256; off <<= 1) {
    float v = (tid >= off) ? lds[tid - off] : 0.f;
    __syncthreads();
    lds[tid] += v;
    __syncthreads();
  }
  float c = (tid == 0) ? 0.f : lds[tid - 1];
#pragma unroll
  for (int i = 0; i < 8; ++i) {
    c += loc[i];
    ur[tid * 8 + i] = loc[i];
    cr[tid * 8 + i] = c;
  }
}

// Monotonic alignment recurrence: one block per (b,h); a_prev in registers.
__global__ __launch_bounds__(256) void k_alpha(const float* __restrict__ pch,
                                               float* __restrict__ alpha) {
  __shared__ float lds[256];
  int bh = blockIdx.x, tid = threadIdx.x;
  float aprev[8];
#pragma unroll
  for (int i = 0; i < 8; ++i) aprev[i] = 0.f;
  if (tid == 0) aprev[0] = 1.f;  // a0 one-hot at s=0
  const float* prow = pch + (size_t)bh * T * S;
  float* arow = alpha + (size_t)bh * T * S;
  for (int t = 0; t < T; ++t) {
    float p[8], ex[8], r[8];
#pragma unroll
    for (int i = 0; i < 8; ++i) p[i] = prow[(size_t)t * S + tid * 8 + i];
    // exclusive cumprod of (1-p): multiplicative block scan
    float tp = 1.f;
#pragma unroll
    for (int i = 0; i < 8; ++i) tp *= (1.f - p[i]);
    lds[tid] = tp;
    __syncthreads();
    for (int off = 1; off < 256; off <<= 1) {
      float v = (tid >= off) ? lds[tid - off] : 1.f;
      __syncthreads();
      lds[tid] *= v;
      __syncthreads();
    }
    float run = (tid == 0) ? 1.f : lds[tid - 1];
    __syncthreads();
    float ts = 0.f;
#pragma unroll
    for (int i = 0; i < 8; ++i) {
      ex[i] = run;
      run *= (1.f - p[i]);
      r[i] = aprev[i] / fmaxf(ex[i], EPS);
      ts += r[i];
    }
    // inclusive cumsum of r
    lds[tid] = ts;
    __syncthreads();
    for (int off = 1; off < 256; off <<= 1) {
      float v = (tid >= off) ? lds[tid - off] : 0.f;
      __syncthreads();
      lds[tid] += v;
      __syncthreads();
    }
    float c = (tid == 0) ? 0.f : lds[tid - 1];
    __syncthreads();
#pragma unroll
    for (int i = 0; i < 8; ++i) {
      c += r[i];
      float at = p[i] * ex[i] * c;
      arow[(size_t)t * S + tid * 8 + i] = at;
      aprev[i] = at;
    }
  }
}

// beta = eu * reverse_inclusive_cumsum(alpha / max(csum,EPS)); beta -> csum buf
__global__ __launch_bounds__(256) void k_beta(const float* __restrict__ alpha,
                                              const float* __restrict__ eu,
                                              float* __restrict__ csum) {
  __shared__ float lds[256];
  size_t row = blockIdx.x;
  int tid = threadIdx.x;
  const float* ar = alpha + row * (size_t)S;
  const float* er = eu + row * (size_t)S;
  float* cr = csum + row * (size_t)S;
  float g[8];
  float ts = 0.f;
#pragma unroll
  for (int i = 0; i < 8; ++i) {
    g[i] = ar[tid * 8 + i] / fmaxf(cr[tid * 8 + i], EPS);
    ts += g[i];
  }
  lds[tid] = ts;
  __syncthreads();
  for (int off = 1; off < 256; off <<= 1) {
    float v = (tid >= off) ? lds[tid - off] : 0.f;
    __syncthreads();
    lds[tid] += v;
    __syncthreads();
  }
  float suf_after = lds[255] - lds[tid];  // sum over threads after tid
  float acc = suf_after, suf[8];
  for (int i = 7; i >= 0; --i) {
    acc += g[i];
    suf[i] = acc;
  }
#pragma unroll
  for (int i = 0; i < 8; ++i) cr[tid * 8 + i] = er[tid * 8 + i] * suf[i];
}

// cv[b,t,h*64+d] = sum_s beta[(b*H+h)*T+t][s] * vp[b*S+s][h*64+d]
// 8 waves split the S reduction; partials reduced through LDS.
__global__ __launch_bounds__(256) void k_av(const float* __restrict__ beta,
                                            const float* __restrict__ vp,
                                            float* __restrict__ cv) {
  __shared__ float op[8 * 16 * 64];
  int tid = threadIdx.x, lane = tid & 31, wave = tid >> 5;
  int l16 = lane & 15, kh = (lane >> 4) * 2;
  int t0 = blockIdx.x * 16, bh = blockIdx.y, bb = bh / H, h = bh % H;
  int kbeg = wave * (S / 8), kend = kbeg + (S / 8);
  const float* arow = beta + ((size_t)bh * T + t0 + l16) * S + kh;
  v8f acc[4] = {};
  for (int k = kbeg; k < kend; k += 4) {
    v2f a;
    a.x = arow[k];
    a.y = arow[k + 1];
    const float* v0 = vp + ((size_t)bb * S + k + kh) * D + h * DH + l16;
#pragma unroll
    for (int j = 0; j < 4; ++j) {
      v2f bf;
      bf.x = v0[j * 16];
      bf.y = v0[j * 16 + D];
      acc[j] = wmma4(a, bf, acc[j]);
    }
  }
  int rbase = (lane >> 4) * 8;
#pragma unroll
  for (int j = 0; j < 4; ++j)
    for (int r = 0; r < 8; ++r)
      op[wave * 1024 + (rbase + r) * 64 + j * 16 + l16] = acc[j][r];
  __syncthreads();
  for (int e = tid * 4; e < tid * 4 + 4; ++e) {
    int rr = e >> 6, nn = e & 63;
    float v = 0.f;
#pragma unroll
    for (int w2 = 0; w2 < 8; ++w2) v += op[w2 * 1024 + rr * 64 + nn];
    cv[((size_t)bb * T + t0 + rr) * D + h * DH + nn] = v;
  }
}

// LayerNorm(x + res) * w + b, one block per row of D=512
__global__ __launch_bounds__(256) void k_ln(const float* __restrict__ x,
                                            const float* __restrict__ res,
                                            const float* __restrict__ w,
                                            const float* __restrict__ bsh,
                                            float* __restrict__ o) {
  __shared__ float lds[256];
  size_t row = blockIdx.x;
  int tid = threadIdx.x;
  float v0 = x[row * D + tid] + (res ? res[row * D + tid] : 0.f);
  float v1 = x[row * D + tid + 256] + (res ? res[row * D + tid + 256] : 0.f);
  lds[tid] = v0 + v1;
  __syncthreads();
  for (int off = 128; off > 0; off >>= 1) {
    if (tid < off) lds[tid] += lds[tid + off];
    __syncthreads();
  }
  float mean = lds[0] / (float)D;
  __syncthreads();
  float d0 = v0 - mean, d1 = v1 - mean;
  lds[tid] = d0 * d0 + d1 * d1;
  __syncthreads();
  for (int off = 128; off > 0; off >>= 1) {
    if (tid < off) lds[tid] += lds[tid + off];
    __syncthreads();
  }
  float rstd = rsqrtf(lds[0] / (float)D + LN_EPS);
  o[row * D + tid] = d0 * rstd * w[tid] + bsh[tid];
  o[row * D + tid + 256] = d1 * rstd * w[tid + 256] + bsh[tid + 256];
}

// ---------------- host orchestration --------------------------------------
static inline void gemm(hipStream_t st, const float* A, const float* W,
                        const float* bias, float* C, int M, int N, int K,
                        int relu) {
  dim3 g(N / 64, M / 128);
  k_gemm<<<g, dim3(256), 0, st>>>(A, W, bias, C, M, N, K, relu);
}

extern "C" void kernel_launch(void* const* d_in, const int* in_sizes, int n_in,
                              void* d_out, int out_size, void* d_ws,
                              size_t ws_size, hipStream_t stream) {
  (void)in_sizes; (void)n_in; (void)out_size; (void)ws_size;
  const float* src = (const float*)d_in[0];
  const float* pos = (const float*)d_in[1];
  const float* tgt = (const float*)d_in[2];
  const float* qpos = (const float*)d_in[3];
  const float* e_qkv_w = (const float*)d_in[4];
  const float* e_qkv_b = (const float*)d_in[5];
  const float* e_out_w = (const float*)d_in[6];
  const float* e_out_b = (const float*)d_in[7];
  const float* e_ff1_w = (const float*)d_in[8];
  const float* e_ff1_b = (const float*)d_in[9];
  const float* e_ff2_w = (const float*)d_in[10];
  const float* e_ff2_b = (const float*)d_in[11];
  const float* e_ln1_w = (const float*)d_in[12];
  const float* e_ln1_b = (const float*)d_in[13];
  const float* e_ln2_w = (const float*)d_in[14];
  const float* e_ln2_b = (const float*)d_in[15];
  const float* d_qkv_w = (const float*)d_in[16];
  const float* d_qkv_b = (const float*)d_in[17];
  const float* d_out_w = (const float*)d_in[18];
  const float* d_out_b = (const float*)d_in[19];
  const float* d_ff1_w = (const float*)d_in[20];
  const float* d_ff1_b = (const float*)d_in[21];
  const float* d_ff2_w = (const float*)d_in[22];
  const float* d_ff2_b = (const float*)d_in[23];
  const float* d_ln1_w = (const float*)d_in[24];
  const float* d_ln1_b = (const float*)d_in[25];
  const float* d_ln2_w = (const float*)d_in[26];
  const float* d_ln2_b = (const float*)d_in[27];
  const float* d_ln3_w = (const float*)d_in[28];
  const float* d_ln3_b = (const float*)d_in[29];
  const float* mkm_w = (const float*)d_in[30];
  const float* mkm_b = (const float*)d_in[31];
  const float* mqm_w = (const float*)d_in[32];
  const float* mqm_b = (const float*)d_in[33];
  const float* mr = (const float*)d_in[34];
  const float* mkc_w = (const float*)d_in[35];
  const float* mkc_b = (const float*)d_in[36];
  const float* mqc_w = (const float*)d_in[37];
  const float* mqc_b = (const float*)d_in[38];
  const float* mv_w = (const float*)d_in[39];
  const float* mv_b = (const float*)d_in[40];
  const float* mo_w = (const float*)d_in[41];
  const float* mo_b = (const float*)d_in[42];
  const float* n_w = (const float*)d_in[43];
  const float* n_b = (const float*)d_in[44];

  // ---- workspace carve ----
  float* ws = (float*)d_ws;
  size_t off = 0;
  auto alloc = [&](size_t n) { float* p = ws + off; off += n; return p; };
  const size_t NS = (size_t)S * B * D;      // 2.10M
  const size_t NT = (size_t)T * B * D;      // 0.52M
  const size_t NB = (size_t)B * H * T * S;  // 16.8M
  float* mem = alloc(NS);
  float* outb = alloc(NT);
  float* xq = alloc(NS);
  float* Qb = alloc(NS);
  float* Kb = alloc(NS);
  float* Vb = alloc(NS);
  float* attnO = alloc(NS);
  float* proj = alloc(NS);
  float* ff1 = alloc((size_t)S * B * FF);
  float* mem_bf = alloc(NS);
  float* key_bf = alloc(NS);
  float* qt = alloc(NT);
  float* qm = alloc(NT);
  float* km = alloc(NS);
  float* qc = alloc(NT);
  float* kc = alloc(NS);
  float* vpb = alloc(NS);
  float* pch = alloc(NB);
  float* ub = alloc(NB);
  float* csb = alloc(NB);
  float* alb = alloc(NB);
  float* cv = alloc(NT);
  float* cvt = alloc(NT);

  auto shb = [](int Sk) {
    return (size_t)(16 * Sk + 256 + 32 + 2048) * sizeof(float);
  };
  auto blocks = [](size_t n) { return (unsigned)((n + 255) / 256); };

  // init activations
  k_add<<<blocks(NS), 256, 0, stream>>>(mem, src, nullptr, (int)NS);
  k_add<<<blocks(NT), 256, 0, stream>>>(outb, tgt, nullptr, (int)NT);

  // ================= encoder =================
  for (int l = 0; l < L; ++l) {
    const float* qkvw = e_qkv_w + (size_t)l * 3 * D * D;
    const float* qkvb = e_qkv_b + (size_t)l * 3 * D;
    k_add<<<blocks(NS), 256, 0, stream>>>(xq, mem, pos, (int)NS);
    gemm(stream, xq, qkvw, qkvb, Qb, S * B, D, D, 0);
    gemm(stream, xq, qkvw + (size_t)D * D, qkvb + D, Kb, S * B, D, D, 0);
    gemm(stream, mem, qkvw + (size_t)2 * D * D, qkvb + 2 * D, Vb, S * B, D, D, 0);
    k_attn<<<dim3(S / 16, H, B), 256, shb(S), stream>>>(Qb, Kb, Vb, attnO, S);
    gemm(stream, attnO, e_out_w + (size_t)l * D * D, e_out_b + (size_t)l * D,
         proj, S * B, D, D, 0);
    k_ln<<<S * B, 256, 0, stream>>>(mem, proj, e_ln1_w + (size_t)l * D,
                                    e_ln1_b + (size_t)l * D, mem);
    gemm(stream, mem, e_ff1_w + (size_t)l * FF * D, e_ff1_b + (size_t)l * FF,
         ff1, S * B, FF, D, 1);
    gemm(stream, ff1, e_ff2_w + (size_t)l * D * FF, e_ff2_b + (size_t)l * D,
         proj, S * B, D, FF, 0);
    k_ln<<<S * B, 256, 0, stream>>>(mem, proj, e_ln2_w + (size_t)l * D,
                                    e_ln2_b + (size_t)l * D, mem);
  }

  // memory in batch-first layouts for MoChA
  k_tr_sb2bs<<<blocks(NS), 256, 0, stream>>>(mem_bf, mem, nullptr, S);
  k_tr_sb2bs<<<blocks(NS), 256, 0, stream>>>(key_bf, mem, pos, S);

  // ================= decoder =================
  for (int l = 0; l < L; ++l) {
    const float* qkvw = d_qkv_w + (size_t)l * 3 * D * D;
    const float* qkvb = d_qkv_b + (size_t)l * 3 * D;
    // --- self attention ---
    k_add<<<blocks(NT), 256, 0, stream>>>(xq, outb, qpos, (int)NT);
    gemm(stream, xq, qkvw, qkvb, Qb, T * B, D, D, 0);
    gemm(stream, xq, qkvw + (size_t)D * D, qkvb + D, Kb, T * B, D, D, 0);
    gemm(stream, outb, qkvw + (size_t)2 * D * D, qkvb + 2 * D, Vb, T * B, D, D, 0);
    k_attn<<<dim3(T / 16, H, B), 256, shb(T), stream>>>(Qb, Kb, Vb, attnO, T);
    gemm(stream, attnO, d_out_w + (size_t)l * D * D, d_out_b + (size_t)l * D,
         proj, T * B, D, D, 0);
    k_ln<<<T * B, 256, 0, stream>>>(outb, proj, d_ln1_w + (size_t)l * D,
                                    d_ln1_b + (size_t)l * D, outb);
    // --- MoChA cross attention ---
    k_tr_sb2bs<<<blocks(NT), 256, 0, stream>>>(qt, outb, qpos, T);
    gemm(stream, qt, mqm_w + (size_t)l * D * D, mqm_b + (size_t)l * D, qm,
         B * T, D, D, 0);
    gemm(stream, key_bf, mkm_w + (size_t)l * D * D, mkm_b + (size_t)l * D, km,
         B * S, D, D, 0);
    gemm(stream, qt, mqc_w + (size_t)l * D * D, mqc_b + (size_t)l * D, qc,
         B * T, D, D, 0);
    gemm(stream, key_bf, mkc_w + (size_t)l * D * D, mkc_b + (size_t)l * D, kc,
         B * S, D, D, 0);
    gemm(stream, mem_bf, mv_w + (size_t)l * D * D, mv_b + (size_t)l * D, vpb,
         B * S, D, D, 0);
    k_qk<<<dim3(S / 64, T / 16, B * H), 32, 0, stream>>>(qm, km, pch, mr, l, 0);
    k_qk<<<dim3(S / 64, T / 16, B * H), 32, 0, stream>>>(qc, kc, ub, nullptr, 0, 1);
    k_expcumsum<<<B * H * T, 256, 0, stream>>>(ub, csb);
    k_alpha<<<B * H, 256, 0, stream>>>(pch, alb);
    k_beta<<<B * H * T, 256, 0, stream>>>(alb, ub, csb);  // beta -> csb
    k_av<<<dim3(T / 16, B * H), 256, 0, stream>>>(csb, vpb, cv);
    gemm(stream, cv, mo_w + (size_t)l * D * D, mo_b + (size_t)l * D, qt,
         B * T, D, D, 0);  // reuse qt as cv-projection
    k_tr_bs2sb<<<blocks(NT), 256, 0, stream>>>(cvt, qt, T);
    k_ln<<<T * B, 256, 0, stream>>>(outb, cvt, d_ln2_w + (size_t)l * D,
                                    d_ln2_b + (size_t)l * D, outb);
    // --- feed forward ---
    gemm(stream, outb, d_ff1_w + (size_t)l * FF * D, d_ff1_b + (size_t)l * FF,
         ff1, T * B, FF, D, 1);
    gemm(stream, ff1, d_ff2_w + (size_t)l * D * FF, d_ff2_b + (size_t)l * D,
         proj, T * B, D, FF, 0);
    k_ln<<<T * B, 256, 0, stream>>>(outb, proj, d_ln3_w + (size_t)l * D,
                                    d_ln3_b + (size_t)l * D, outb);
  }

  // final norm -> output [T,B,D]
  k_ln<<<T * B, 256, 0, stream>>>(outb, nullptr, n_w, n_b, (float*)d_out);
}